// GNN_65240553226519
// MI455X (gfx1250) — compile-verified
//
#include <hip/hip_runtime.h>
#include <cmath>

#define NN 50000
#define NE 800000
#define NG 256

typedef __attribute__((ext_vector_type(2))) float v2f;
typedef __attribute__((ext_vector_type(8))) float v8f;

// Hardware f32 atomic add (non-returning, device scope) — guarantees
// global_atomic_add_f32 instead of a CAS loop. Tracked on STOREcnt;
// S_ENDPGM's implicit wait-idle covers completion before kernel end.
__device__ __forceinline__ void atom_add_f32(float* p, float v) {
  asm volatile("global_atomic_add_f32 %0, %1, off scope:SCOPE_DEV"
               :: "v"(p), "v"(v) : "memory");
}

__global__ void zero_kernel(float* __restrict__ p, int n4) {
  int stride = gridDim.x * blockDim.x;
  float4 z = make_float4(0.f, 0.f, 0.f, 0.f);
  for (int i = blockIdx.x * blockDim.x + threadIdx.x; i < n4; i += stride)
    ((float4*)p)[i] = z;
}

// One edge handled by D/4 consecutive threads -> coalesced float4 gather,
// 4 hw f32 atomics into the destination row (L2-resident: agg <= 25.6MB).
template<int D>
__global__ void scatter_kernel(const float* __restrict__ feat,
                               const int* __restrict__ src,
                               const int* __restrict__ dst,
                               const float* __restrict__ ew,
                               float* __restrict__ agg) {
  constexpr int C = D / 4;
  int tid = blockIdx.x * blockDim.x + threadIdx.x;
  if (tid >= NE * C) return;
  int e = tid / C;
  int c = (tid % C) * 4;
  int s = src[e], d = dst[e];
  float w = ew[e];
  const float4 f = *(const float4*)(feat + (size_t)s * D + c);
  float* base = agg + (size_t)d * D + c;
  atom_add_f32(base + 0, f.x * w);
  atom_add_f32(base + 1, f.y * w);
  atom_add_f32(base + 2, f.z * w);
  atom_add_f32(base + 3, f.w * w);
}

// Out = act( [addm] + bias + A1 @ W1^T [ + A2 @ W2^T ] )
// A* : [NN, K] row-major, W* : [OUT, K] row-major. One wave per 16-node tile,
// full OUT width (NT fragments), fp32 WMMA 16x16x4.
template<int K, int OUT, bool SECOND, bool ADDM, bool ACT>
__global__ __launch_bounds__(256) void node_gemm_kernel(
    const float* __restrict__ A1, const float* __restrict__ W1,
    const float* __restrict__ A2, const float* __restrict__ W2,
    const float* __restrict__ bias, const float* __restrict__ addm,
    float* __restrict__ Out) {
  constexpr int NT = OUT / 16;
  const int wave = blockIdx.x * (blockDim.x >> 5) + (threadIdx.x >> 5);
  if (wave >= NN / 16) return;          // 50000 = 3125 * 16, exact
  const int lane = threadIdx.x & 31;
  const int col  = lane & 15;           // M row for A loads, N col for B/C
  const int hi   = lane >> 4;           // K offset select (0 -> K,K+1 ; 1 -> K+2,K+3)
  const int m0   = wave * 16;

  v8f acc[NT];
  #pragma unroll
  for (int t = 0; t < NT; t++) {
    float b = bias ? bias[t * 16 + col] : 0.0f;   // C layout: col = lane&15
    #pragma unroll
    for (int r = 0; r < 8; r++) acc[t][r] = b;
  }
  if constexpr (ADDM) {
    #pragma unroll
    for (int t = 0; t < NT; t++) {
      #pragma unroll
      for (int r = 0; r < 8; r++)       // C layout row: m = m0 + r + hi*8
        acc[t][r] += addm[(size_t)(m0 + hi * 8 + r) * OUT + t * 16 + col];
    }
  }

  {
    const float* arow = A1 + (size_t)(m0 + col) * K + hi * 2;
    #pragma unroll 4
    for (int k = 0; k < K; k += 4) {
      v2f a; a.x = arow[k]; a.y = arow[k + 1];
      #pragma unroll
      for (int t = 0; t < NT; t++) {
        const float* wrow = W1 + (size_t)(t * 16 + col) * K + hi * 2;
        v2f b; b.x = wrow[k]; b.y = wrow[k + 1];
        acc[t] = __builtin_amdgcn_wmma_f32_16x16x4_f32(
            false, a, false, b, (short)0, acc[t], false, false);
      }
    }
  }
  if constexpr (SECOND) {
    const float* arow = A2 + (size_t)(m0 + col) * K + hi * 2;
    #pragma unroll 4
    for (int k = 0; k < K; k += 4) {
      v2f a; a.x = arow[k]; a.y = arow[k + 1];
      #pragma unroll
      for (int t = 0; t < NT; t++) {
        const float* wrow = W2 + (size_t)(t * 16 + col) * K + hi * 2;
        v2f b; b.x = wrow[k]; b.y = wrow[k + 1];
        acc[t] = __builtin_amdgcn_wmma_f32_16x16x4_f32(
            false, a, false, b, (short)0, acc[t], false, false);
      }
    }
  }

  #pragma unroll
  for (int r = 0; r < 8; r++) {
    const int m = m0 + hi * 8 + r;
    #pragma unroll
    for (int t = 0; t < NT; t++) {
      float v = acc[t][r];
      if constexpr (ACT) v = tanhf(v);
      Out[(size_t)m * OUT + t * 16 + col] = v;
    }
  }
}

// global add pool: g[batch[n]] += h3[n]  (h3 is [NN, 64])
__global__ void pool_kernel(const float* __restrict__ h,
                            const int* __restrict__ batch,
                            float* __restrict__ g) {
  int tid = blockIdx.x * blockDim.x + threadIdx.x;
  if (tid >= NN * 16) return;
  int n = tid >> 4;
  int c = (tid & 15) * 4;
  int b = batch[n];
  const float4 f = *(const float4*)(h + (size_t)n * 64 + c);
  float* base = g + (size_t)b * 64 + c;
  atom_add_f32(base + 0, f.x);
  atom_add_f32(base + 1, f.y);
  atom_add_f32(base + 2, f.z);
  atom_add_f32(base + 3, f.w);
}

// 256 graphs x (64->32 relu ->16 relu ->1): one thread per graph.
__global__ void mlp_kernel(const float* __restrict__ g,
                           const float* __restrict__ Wm1, const float* __restrict__ bm1,
                           const float* __restrict__ Wm2, const float* __restrict__ bm2,
                           const float* __restrict__ Wm3, const float* __restrict__ bm3,
                           float* __restrict__ out) {
  int gi = blockIdx.x * blockDim.x + threadIdx.x;
  if (gi >= NG) return;
  const float* row = g + (size_t)gi * 64;
  float h1[32];
  for (int i = 0; i < 32; i++) {
    float s = bm1[i];
    const float* w = Wm1 + i * 64;
    for (int k = 0; k < 64; k++) s = fmaf(row[k], w[k], s);
    h1[i] = fmaxf(s, 0.f);
  }
  float h2[16];
  for (int i = 0; i < 16; i++) {
    float s = bm2[i];
    const float* w = Wm2 + i * 32;
    for (int k = 0; k < 32; k++) s = fmaf(h1[k], w[k], s);
    h2[i] = fmaxf(s, 0.f);
  }
  float o = bm3[0];
  for (int k = 0; k < 16; k++) o = fmaf(h2[k], Wm3[k], o);
  out[gi] = o;
}

extern "C" void kernel_launch(void* const* d_in, const int* in_sizes, int n_in,
                              void* d_out, int out_size, void* d_ws, size_t ws_size,
                              hipStream_t stream) {
  const float* x      = (const float*)d_in[0];
  const int*   ei     = (const int*)  d_in[1];
  const int*   batch  = (const int*)  d_in[2];
  const float* ew     = (const float*)d_in[3];
  const float* W1r    = (const float*)d_in[4];
  const float* b1     = (const float*)d_in[5];
  const float* W1root = (const float*)d_in[6];
  const float* W2r    = (const float*)d_in[7];
  const float* b2     = (const float*)d_in[8];
  const float* W2root = (const float*)d_in[9];
  const float* W3r    = (const float*)d_in[10];
  const float* b3     = (const float*)d_in[11];
  const float* W3root = (const float*)d_in[12];
  const float* Wm1    = (const float*)d_in[13];
  const float* bm1    = (const float*)d_in[14];
  const float* Wm2    = (const float*)d_in[15];
  const float* bm2    = (const float*)d_in[16];
  const float* Wm3    = (const float*)d_in[17];
  const float* bm3    = (const float*)d_in[18];
  const int* src = ei;
  const int* dst = ei + NE;
  float* out = (float*)d_out;

  char* ws = (char*)d_ws;
  float* agg  = (float*)(ws);                 // 50000*128*4 = 25.6 MB (reused)
  float* hA   = (float*)(ws + 25600000);      // h1, then t3, then h3
  float* hB   = (float*)(ws + 51200000);      // h2
  float* gbuf = (float*)(ws + 76800000);      // 256*64*4

  const int GEMM_BLOCKS = (3125 + 7) / 8;     // 8 waves (16-node tiles) / 256-thread block

  // ---- Layer 1: agg64 = segsum(x[src]*w); h1 = tanh(agg@W1r^T + x@W1root^T + b1)
  zero_kernel<<<2048, 256, 0, stream>>>(agg, NN * 64 / 4);
  scatter_kernel<64><<<(NE * 16 + 255) / 256, 256, 0, stream>>>(x, src, dst, ew, agg);
  node_gemm_kernel<64, 128, true, false, true><<<GEMM_BLOCKS, 256, 0, stream>>>(
      agg, W1r, x, W1root, b1, nullptr, hA);

  // ---- Layer 2: agg128 = segsum(h1[src]*w); h2 = tanh(agg@W2r^T + h1@W2root^T + b2)
  zero_kernel<<<2048, 256, 0, stream>>>(agg, NN * 128 / 4);
  scatter_kernel<128><<<(NE * 32 + 255) / 256, 256, 0, stream>>>(hA, src, dst, ew, agg);
  node_gemm_kernel<128, 128, true, false, true><<<GEMM_BLOCKS, 256, 0, stream>>>(
      agg, W2r, hA, W2root, b2, nullptr, hB);

  // ---- Layer 3 (transform-first: segsum(h2[src]*w)@W3r^T == segsum((h2@W3r^T)[src]*w),
  //      halves atomic traffic by scattering in 64-d instead of 128-d)
  node_gemm_kernel<128, 64, false, false, false><<<GEMM_BLOCKS, 256, 0, stream>>>(
      hB, W3r, nullptr, nullptr, nullptr, nullptr, hA);                  // t3 = h2@W3r^T
  zero_kernel<<<2048, 256, 0, stream>>>(agg, NN * 64 / 4);
  scatter_kernel<64><<<(NE * 16 + 255) / 256, 256, 0, stream>>>(hA, src, dst, ew, agg);
  node_gemm_kernel<128, 64, false, true, true><<<GEMM_BLOCKS, 256, 0, stream>>>(
      hB, W3root, nullptr, nullptr, b3, agg, hA);                        // h3

  // ---- Pool + MLP head
  zero_kernel<<<64, 256, 0, stream>>>(gbuf, NG * 64 / 4);
  pool_kernel<<<(NN * 16 + 255) / 256, 256, 0, stream>>>(hA, batch, gbuf);
  mlp_kernel<<<1, 256, 0, stream>>>(gbuf, Wm1, bm1, Wm2, bm2, Wm3, bm3, out);
}